// model_25451976196817
// MI455X (gfx1250) — compile-verified
//
#include <hip/hip_runtime.h>
#include <hip/hip_bf16.h>

// ---------------------------------------------------------------------------
// GNN inference for MI455X (gfx1250, wave32).
// Memory-bound workload: dominated by the E=1.2M-edge gather/scatter with 64
// channels (~600MB round trip; scatter target fits in 192MB L2). Dense GEMMs
// (N x 3 -> 64 and N x 64 -> 50) are tiny and run on V_WMMA_F32_16X16X4_F32
// so they stay bit-compatible with the f32 reference while exercising the
// CDNA5 matrix path.
// ---------------------------------------------------------------------------

#define N_NODES 100000
#define N_EDGES 1200000
#define N_GRAPHS 1000
#define BN_EPS 1e-5f

typedef __attribute__((ext_vector_type(2))) float v2f;
typedef __attribute__((ext_vector_type(8))) float v8f;

// ---------------- utility ----------------

__global__ void zero_kernel(float* p, long n) {
    long i = (long)blockIdx.x * blockDim.x + threadIdx.x;
    long stride = (long)gridDim.x * blockDim.x;
    for (; i < n; i += stride) p[i] = 0.0f;
}

// ---------------- degrees ----------------

__global__ void degrees_kernel(const int* __restrict__ src, const int* __restrict__ dst,
                               float* deg_out, float* deg_in) {
    int e = blockIdx.x * blockDim.x + threadIdx.x;
    if (e >= N_EDGES) return;
    atomicAdd(&deg_out[src[e]], 1.0f);
    atomicAdd(&deg_in[dst[e]], 1.0f);
}

__global__ void rsqrt_deg_kernel(float* deg_out, float* deg_in) {
    int i = blockIdx.x * blockDim.x + threadIdx.x;
    if (i >= N_NODES) return;
    deg_out[i] = rsqrtf(fmaxf(deg_out[i], 1.0f));
    deg_in[i]  = rsqrtf(fmaxf(deg_in[i], 1.0f));
}

// ---------------- conv1: edge scatter (3 channels, stride-4 padded) ----------------

__global__ void conv1_scatter(const int* __restrict__ src, const int* __restrict__ dst,
                              const int* __restrict__ labels, const float* __restrict__ emb,
                              const float* __restrict__ dinv_out, float* m1) {
    int e = blockIdx.x * blockDim.x + threadIdx.x;
    if (e >= N_EDGES) return;
    int s = src[e], d = dst[e];
    float sc = dinv_out[s];
    const float* er = emb + labels[s] * 3;
    atomicAdd(&m1[d * 4 + 0], er[0] * sc);
    atomicAdd(&m1[d * 4 + 1], er[1] * sc);
    atomicAdd(&m1[d * 4 + 2], er[2] * sc);
    // m1[d*4+3] stays 0 -> zero pad for WMMA K=4
}

// ---------------- conv1: node GEMM [N,4] @ [4,64] via WMMA f32 16x16x4 ----------------
// block = (32,4): 4 waves, wave w computes output column tile w (cols w*16..w*16+15).

__global__ void conv1_node(const float* __restrict__ m1, const float* __restrict__ dinv_in,
                           const float* __restrict__ W1, const float* __restrict__ b1,
                           float* __restrict__ h1) {
    int rowTile = blockIdx.x;            // 6250 tiles of 16 rows, N divides exactly
    int wave = threadIdx.y;              // column tile 0..3
    int lane = threadIdx.x;              // 0..31
    int r = lane & 15;
    int hi = lane >> 4;                  // lanes 16..31 hold K=2..3 / rows M+8
    int row = rowTile * 16 + r;
    int col = wave * 16 + r;
    int k0 = hi * 2;

    float dinv = dinv_in[row];
    v2f a;
    a.x = m1[row * 4 + k0] * dinv;
    a.y = m1[row * 4 + k0 + 1] * dinv;   // k==3 is the zero pad column

    v2f b;                               // W1 is [3,64] row-major; K row 3 is zero
    b.x = (k0 < 3)     ? W1[k0 * 64 + col]       : 0.0f;
    b.y = (k0 + 1 < 3) ? W1[(k0 + 1) * 64 + col] : 0.0f;

    v8f c;
    float bias = b1[col];
    #pragma unroll
    for (int v = 0; v < 8; ++v) c[v] = bias;

    c = __builtin_amdgcn_wmma_f32_16x16x4_f32(false, a, false, b, (short)0, c, false, false);

    #pragma unroll
    for (int v = 0; v < 8; ++v) {
        int M = v + hi * 8;
        h1[(long)(rowTile * 16 + M) * 64 + col] = c[v];
    }
}

// ---------------- batch-norm stats (sum, sum of squares per channel) ----------------
// block = (64,4); grid-stride over rows; block-reduce then atomics.

__global__ void bn_stats(const float* __restrict__ h, int ncols, float* sum, float* sq) {
    int c = threadIdx.x;                 // 0..63 (stride is always 64)
    float s = 0.0f, s2 = 0.0f;
    if (c < ncols) {
        for (long row = (long)blockIdx.x * blockDim.y + threadIdx.y; row < N_NODES;
             row += (long)gridDim.x * blockDim.y) {
            float v = h[row * 64 + c];
            s += v; s2 += v * v;
        }
    }
    __shared__ float ls[4][64], ls2[4][64];
    ls[threadIdx.y][c] = s;
    ls2[threadIdx.y][c] = s2;
    __syncthreads();
    if (threadIdx.y == 0 && c < ncols) {
        float t = 0.0f, t2 = 0.0f;
        #pragma unroll
        for (int y = 0; y < 4; ++y) { t += ls[y][c]; t2 += ls2[y][c]; }
        atomicAdd(&sum[c], t);
        atomicAdd(&sq[c], t2);
    }
}

// ---------------- BN1 + ReLU + rsqrt(deg_out) scale, in place ----------------

__global__ void bn1_apply(float* h1, const float* __restrict__ sum, const float* __restrict__ sq,
                          const float* __restrict__ gamma, const float* __restrict__ beta,
                          const float* __restrict__ dinv_out) {
    int idx = blockIdx.x * blockDim.x + threadIdx.x;   // N*64 threads
    if (idx >= N_NODES * 64) return;
    int c = idx & 63;
    int row = idx >> 6;
    float mean = sum[c] * (1.0f / N_NODES);
    float var = sq[c] * (1.0f / N_NODES) - mean * mean;
    float v = (h1[idx] - mean) * rsqrtf(var + BN_EPS) * gamma[c] + beta[c];
    h1[idx] = fmaxf(v, 0.0f) * dinv_out[row];
}

// ---------------- conv2: edge scatter, thread per (edge, channel) ----------------
// 64 consecutive threads share one edge: coalesced 256B row read, 256B atomic target.

__global__ void conv2_scatter(const int* __restrict__ src, const int* __restrict__ dst,
                              const float* __restrict__ h1s, float* m2) {
    int idx = blockIdx.x * blockDim.x + threadIdx.x;
    if (idx >= N_EDGES * 64) return;
    int c = idx & 63;
    int e = idx >> 6;
    atomicAdd(&m2[(long)dst[e] * 64 + c], h1s[(long)src[e] * 64 + c]);
}

// ---------------- conv2: node GEMM [N,64] @ [64,50] via 16 chained WMMA ----------------

__global__ void conv2_node(const float* __restrict__ m2, const float* __restrict__ dinv_in,
                           const float* __restrict__ W2, const float* __restrict__ b2,
                           float* __restrict__ h2) {
    int rowTile = blockIdx.x;
    int wave = threadIdx.y;              // column tile 0..3 (cols 48..63 padded)
    int lane = threadIdx.x;
    int r = lane & 15;
    int hi = lane >> 4;
    int row = rowTile * 16 + r;
    int col = wave * 16 + r;
    bool cok = col < 50;

    float dinv = dinv_in[row];
    const float* arow = m2 + (long)row * 64;

    v8f c;
    float bias = cok ? b2[col] : 0.0f;
    #pragma unroll
    for (int v = 0; v < 8; ++v) c[v] = bias;

    #pragma unroll
    for (int kk = 0; kk < 16; ++kk) {
        int k0 = kk * 4 + hi * 2;
        v2f a;
        a.x = arow[k0] * dinv;
        a.y = arow[k0 + 1] * dinv;
        v2f b;                            // W2 is [64,50] row-major
        b.x = cok ? W2[k0 * 50 + col] : 0.0f;
        b.y = cok ? W2[(k0 + 1) * 50 + col] : 0.0f;
        c = __builtin_amdgcn_wmma_f32_16x16x4_f32(false, a, false, b, (short)0, c, false, false);
    }

    if (cok) {
        #pragma unroll
        for (int v = 0; v < 8; ++v) {
            int M = v + hi * 8;
            h2[(long)(rowTile * 16 + M) * 64 + col] = c[v];
        }
    }
}

// ---------------- per-graph count + mean readout (with BN2 applied on the fly) ------

__global__ void graph_count(const int* __restrict__ graph_ids, float* gcnt) {
    int i = blockIdx.x * blockDim.x + threadIdx.x;
    if (i >= N_NODES) return;
    atomicAdd(&gcnt[graph_ids[i]], 1.0f);
}

__global__ void readout_scatter(const float* __restrict__ h2, const float* __restrict__ sum,
                                const float* __restrict__ sq, const float* __restrict__ gamma,
                                const float* __restrict__ beta, const int* __restrict__ graph_ids,
                                float* gsum) {
    int idx = blockIdx.x * blockDim.x + threadIdx.x;   // N*64 threads, channels >=50 idle
    if (idx >= N_NODES * 64) return;
    int c = idx & 63;
    int row = idx >> 6;
    if (c >= 50) return;
    float mean = sum[c] * (1.0f / N_NODES);
    float var = sq[c] * (1.0f / N_NODES) - mean * mean;
    float v = (h2[(long)row * 64 + c] - mean) * rsqrtf(var + BN_EPS) * gamma[c] + beta[c];
    atomicAdd(&gsum[graph_ids[row] * 50 + c], v);
}

// ---------------- heads: x, geuristic, y, final (one block per graph) ----------------
// d_out layout (return order): x[G*50] | final[G] | y[G*50] | geuristic[G*50]

__global__ void head_kernel(const float* __restrict__ gsum, const float* __restrict__ gcnt,
                            const float* __restrict__ Wb, const float* __restrict__ bb,
                            const float* __restrict__ Wfc, const float* __restrict__ bfc,
                            const float* __restrict__ Wfc2, const float* __restrict__ bfc2,
                            float* out) {
    int g = blockIdx.x;
    int t = threadIdx.x;                 // 64 threads
    __shared__ float xr[50], yr[50];
    float cnt = fmaxf(gcnt[g], 1.0f);
    if (t < 50) {
        float x = gsum[g * 50 + t] / cnt;
        xr[t] = x;
        out[g * 50 + t] = x;                               // x
    }
    __syncthreads();
    if (t < 50) {
        float geur = bb[t];
        float yv = bfc[t];
        for (int k = 0; k < 50; ++k) {
            float x = xr[k];
            geur += x * Wb[k * 50 + t];
            yv += x * Wfc[k * 50 + t];
        }
        yv -= geur;
        yr[t] = yv;
        out[(N_GRAPHS * 51) + g * 50 + t] = yv;            // y      @ 51000
        out[(N_GRAPHS * 101) + g * 50 + t] = geur;         // geur   @ 101000
    }
    __syncthreads();
    if (t == 0) {
        float f = bfc2[0];
        for (int k = 0; k < 50; ++k) f += yr[k] * Wfc2[k];
        out[N_GRAPHS * 50 + g] = f;                        // final  @ 50000
    }
}

// ---------------------------------------------------------------------------

extern "C" void kernel_launch(void* const* d_in, const int* in_sizes, int n_in,
                              void* d_out, int out_size, void* d_ws, size_t ws_size,
                              hipStream_t stream) {
    const int* node_labels = (const int*)d_in[0];
    const int* edge_src    = (const int*)d_in[1];
    const int* edge_dst    = (const int*)d_in[2];
    const int* graph_ids   = (const int*)d_in[3];
    const float* emb  = (const float*)d_in[4];
    const float* W1   = (const float*)d_in[5];
    const float* b1   = (const float*)d_in[6];
    const float* g1   = (const float*)d_in[7];
    const float* be1  = (const float*)d_in[8];
    const float* W2   = (const float*)d_in[9];
    const float* b2   = (const float*)d_in[10];
    const float* g2   = (const float*)d_in[11];
    const float* be2  = (const float*)d_in[12];
    const float* Wb   = (const float*)d_in[13];
    const float* bb   = (const float*)d_in[14];
    const float* Wfc  = (const float*)d_in[15];
    const float* bfc  = (const float*)d_in[16];
    const float* Wfc2 = (const float*)d_in[17];
    const float* bfc2 = (const float*)d_in[18];
    float* out = (float*)d_out;

    const long N = N_NODES;
    float* ws = (float*)d_ws;
    float* dinv_out = ws;                 // N   (degree -> rsqrt in place)
    float* dinv_in  = ws + N;             // N
    float* m1       = ws + 2 * N;         // 4N  (stride 4, col 3 = zero pad)
    float* h1       = ws + 6 * N;         // 64N (h1 raw -> h1 scaled -> reused as h2)
    float* m2       = ws + 70 * N;        // 64N
    float* bn1_sum  = ws + 134 * N;       // 64
    float* bn1_sq   = bn1_sum + 64;       // 64
    float* bn2_sum  = bn1_sum + 128;      // 64
    float* bn2_sq   = bn1_sum + 192;      // 64
    float* gsum     = bn1_sum + 256;      // 50*G
    float* gcnt     = gsum + 50 * N_GRAPHS; // G
    float* h2       = h1;                 // conv2 output reuses h1's buffer (stride 64)

    // 1) zero all accumulation regions (h1 is fully overwritten, no zeroing needed)
    zero_kernel<<<2048, 256, 0, stream>>>(dinv_out, 6 * N);
    zero_kernel<<<2048, 256, 0, stream>>>(m2, 64 * N + 256 + 51 * N_GRAPHS);

    // 2) degrees -> rsqrt(clip(deg,1))
    degrees_kernel<<<(N_EDGES + 255) / 256, 256, 0, stream>>>(edge_src, edge_dst, dinv_out, dinv_in);
    rsqrt_deg_kernel<<<(N_NODES + 255) / 256, 256, 0, stream>>>(dinv_out, dinv_in);

    // 3) conv1 edge phase
    conv1_scatter<<<(N_EDGES + 255) / 256, 256, 0, stream>>>(edge_src, edge_dst, node_labels,
                                                             emb, dinv_out, m1);

    // 4) conv1 node GEMM (WMMA f32 16x16x4)
    conv1_node<<<N_NODES / 16, dim3(32, 4), 0, stream>>>(m1, dinv_in, W1, b1, h1);

    // 5) BN1 stats -> fused BN + ReLU + deg_out scale (in place)
    bn_stats<<<256, dim3(64, 4), 0, stream>>>(h1, 64, bn1_sum, bn1_sq);
    bn1_apply<<<(N_NODES * 64 + 255) / 256, 256, 0, stream>>>(h1, bn1_sum, bn1_sq, g1, be1, dinv_out);

    // 6) conv2 edge phase (dominant memory cost)
    conv2_scatter<<<(N_EDGES * 64 + 255) / 256, 256, 0, stream>>>(edge_src, edge_dst, h1, m2);

    // 7) conv2 node GEMM (16 chained WMMA f32 16x16x4, K=64)
    conv2_node<<<N_NODES / 16, dim3(32, 4), 0, stream>>>(m2, dinv_in, W2, b2, h2);

    // 8) BN2 stats -> per-graph mean readout
    bn_stats<<<256, dim3(64, 4), 0, stream>>>(h2, 50, bn2_sum, bn2_sq);
    graph_count<<<(N_NODES + 255) / 256, 256, 0, stream>>>(graph_ids, gcnt);
    readout_scatter<<<(N_NODES * 64 + 255) / 256, 256, 0, stream>>>(h2, bn2_sum, bn2_sq, g2, be2,
                                                                    graph_ids, gsum);

    // 9) heads
    head_kernel<<<N_GRAPHS, 64, 0, stream>>>(gsum, gcnt, Wb, bb, Wfc, bfc, Wfc2, bfc2, out);
}